// STRG_28398323761232
// MI455X (gfx1250) — compile-verified
//
#include <hip/hip_runtime.h>

typedef float v2f  __attribute__((ext_vector_type(2)));
typedef float v8f  __attribute__((ext_vector_type(8)));
typedef unsigned int u32x4 __attribute__((ext_vector_type(4)));
typedef int   i32x4 __attribute__((ext_vector_type(4)));
typedef int   i32x8 __attribute__((ext_vector_type(8)));

#define NCC  64
#define NB   32
#define TT   4000
#define NFB  5
#define PADL 27
#define TEXTN (TT + 2*PADL)      // 4054
#define NN   (NCC*NFB)           // 320
#define NCH  (NFB*NB*NCC)        // 10240
#define NCHUNK ((TT + 63)/64)    // 63

// ---------------- forward IIR over odd-padded extension ----------------
__global__ void filt_forward(const float* __restrict__ x_in,
                             const float* __restrict__ bco,
                             const float* __restrict__ aco,
                             const float* __restrict__ zico,
                             float* __restrict__ filtExt)
{
    int ch = blockIdx.x * blockDim.x + threadIdx.x;
    if (ch >= NCH) return;
    int f   = ch / (NB*NCC);
    int rem = ch - f*(NB*NCC);
    int bb  = rem / NCC;
    int c   = rem - bb*NCC;
    const float* x = x_in + ((size_t)bb*NCC + c) * TT;

    float bc[9], ac[9], zi[8];
#pragma unroll
    for (int i = 0; i < 9; ++i) { bc[i] = bco[f*9+i]; ac[i] = aco[f*9+i]; }
#pragma unroll
    for (int i = 0; i < 8; ++i) zi[i] = zico[f*8+i];

    float x0 = x[0], xl = x[TT-1];
    float ext0 = 2.f*x0 - x[PADL];
    float z[8];
#pragma unroll
    for (int i = 0; i < 8; ++i) z[i] = zi[i]*ext0;

    float* yrow = filtExt + (size_t)ch * TEXTN;
    for (int t = 0; t < TEXTN; ++t) {
        float xt;
        if (t < PADL)           xt = 2.f*x0 - x[PADL - t];
        else if (t < PADL+TT)   xt = x[t - PADL];
        else                    xt = 2.f*xl - x[TT-2 - (t - (PADL+TT))];
        float y = fmaf(bc[0], xt, z[0]);
#pragma unroll
        for (int i = 0; i < 7; ++i)
            z[i] = z[i+1] + bc[i+1]*xt - ac[i+1]*y;
        z[7] = bc[8]*xt - ac[8]*y;
        yrow[t] = y;
    }
}

// ---- backward IIR (in place, reversed) + center in place + stats ----
__global__ void filt_backward(const float* __restrict__ bco,
                              const float* __restrict__ aco,
                              const float* __restrict__ zico,
                              float* __restrict__ filtExt,
                              float* __restrict__ ssBuf,
                              float* __restrict__ nodef,
                              float* __restrict__ bandp)
{
    int ch = blockIdx.x * blockDim.x + threadIdx.x;
    if (ch >= NCH) return;
    int f   = ch / (NB*NCC);
    int rem = ch - f*(NB*NCC);
    int bb  = rem / NCC;
    int c   = rem - bb*NCC;

    float bc[9], ac[9], zi[8];
#pragma unroll
    for (int i = 0; i < 9; ++i) { bc[i] = bco[f*9+i]; ac[i] = aco[f*9+i]; }
#pragma unroll
    for (int i = 0; i < 8; ++i) zi[i] = zico[f*8+i];

    float* yrow = filtExt + (size_t)ch * TEXTN;
    float y0 = yrow[TEXTN-1];
    float z[8];
#pragma unroll
    for (int i = 0; i < 8; ++i) z[i] = zi[i]*y0;

    float sum = 0.f, sumsq = 0.f;
    for (int s = 0; s < TEXTN; ++s) {
        int pos = TEXTN-1-s;
        float xt = yrow[pos];
        float y = fmaf(bc[0], xt, z[0]);
#pragma unroll
        for (int i = 0; i < 7; ++i)
            z[i] = z[i+1] + bc[i+1]*xt - ac[i+1]*y;
        z[7] = bc[8]*xt - ac[8]*y;
        yrow[pos] = y;               // final[t] lands at pos = PADL + t
        if (pos >= PADL && pos < PADL+TT) {
            sum   += y;
            sumsq  = fmaf(y, y, sumsq);
        }
    }
    float mean = sum * (1.f/TT);
    float ssc  = sumsq - sum*sum*(1.f/TT);   // sum of squares of centered
    // center in place (row is L2-resident) so the GEMM stages raw tiles
    for (int t = 0; t < TT; ++t)
        yrow[PADL + t] -= mean;

    ssBuf[ch] = ssc;
    float bp = ssc * (1.f/TT);
    nodef[(size_t)bb*NN + f*NCC + c]      = bp;   // (B, NF*NC, 1)
    bandp[(size_t)bb*NCC*NFB + c*NFB + f] = bp;   // (B, NC, NF)
}

// -------- fill A with 0.5*A_spatial (off-diag blocks keep this) --------
__global__ void init_A(const float* __restrict__ Asp, float* __restrict__ Aout)
{
    int idx = blockIdx.x*blockDim.x + threadIdx.x;
    const int total = NB*NN*NN;
    if (idx >= total) return;
    int ij = idx % (NN*NN);
    Aout[idx] = 0.5f * Asp[ij];
}

// -------- batched 64x64xK=4000 covariance: TDM staging + f32 WMMA ------
__global__ __launch_bounds__(512)
void cov_wmma(const float* __restrict__ filtExt,
              const float* __restrict__ ssBuf,
              const float* __restrict__ Asp,
              float* __restrict__ Aout)
{
    __shared__ float sh[2][NCC][68];              // 68-pitch via TDM padding
    int fb = blockIdx.x;                          // 0..159
    int f  = fb / NB;
    int bb = fb % NB;
    int chbase = (f*NB + bb) * NCC;
    const float* base = filtExt + (size_t)chbase * TEXTN + PADL;

    int tid  = threadIdx.x;
    int wave = tid >> 5;                          // 16 waves -> 16 tiles
    int lane = tid & 31;
    int half = lane >> 4;                         // K sub-pair select
    int r    = lane & 15;
    int m0   = (wave >> 2) * 16;
    int n0   = (wave & 3)  * 16;

#if __has_builtin(__builtin_amdgcn_tensor_load_to_lds)
    unsigned long long gaddr0 = (unsigned long long)(uintptr_t)base;
    unsigned lds0 = (unsigned)(uintptr_t)&sh[0][0][0];
    unsigned lds1 = (unsigned)(uintptr_t)&sh[1][0][0];

    auto issue_tdm = [&](int chunk) {
        unsigned long long ga = gaddr0 + (unsigned long long)chunk * 256ull; // 64 f32
        unsigned tdim0 = (unsigned)(TT - chunk*64);  // remaining K; OOB -> zero-fill
        u32x4 g0;
        g0.x = 1u;                                   // count=1, normal mode
        g0.y = (chunk & 1) ? lds1 : lds0;            // lds_addr
        g0.z = (unsigned)(ga & 0xFFFFFFFFu);         // global_addr[31:0]
        g0.w = (unsigned)((ga >> 32) & 0x01FFFFFFu)  // global_addr[56:32]
             | (2u << 30);                           // type = 2 (image)
        i32x8 g1;
        g1[0] = (int)((2u<<16)                       // data_size = 4B
                    | (1u<<20)                       // pad_enable
                    | (5u<<22)                       // pad_interval: 64 DWORDs
                    | (3u<<25));                     // pad_amount: 4 DWORDs -> pitch 68
        g1[1] = (int)((tdim0 & 0xFFFFu) << 16);      // tensor_dim0 lo
        g1[2] = (int)((tdim0 >> 16) | (64u << 16));  // tensor_dim0 hi | tensor_dim1 lo
        g1[3] = (int)(64u << 16);                    // tensor_dim1 hi | tile_dim0 = 64
        g1[4] = 64;                                  // tile_dim1 = 64, tile_dim2 = 0
        g1[5] = TEXTN;                               // tensor_dim0_stride = 4054
        g1[6] = 0;
        g1[7] = 0;
        i32x4 z4 = {0, 0, 0, 0};
        i32x8 z8 = {0, 0, 0, 0, 0, 0, 0, 0};
        __builtin_amdgcn_tensor_load_to_lds(g0, g1, z4, z4, z8, 0);
    };

    if (tid < 32) issue_tdm(0);

    v8f acc = {};
    for (int c = 0; c < NCHUNK; ++c) {
        if (tid < 32) {
            if (c + 1 < NCHUNK) { issue_tdm(c + 1);  // prefetch next tile
                                  __builtin_amdgcn_s_wait_tensorcnt(1); }
            else                  __builtin_amdgcn_s_wait_tensorcnt(0);
        }
        __syncthreads();                             // tile c visible to all waves
        const float (*buf)[68] = sh[c & 1];
#pragma unroll
        for (int kk = 0; kk < 16; ++kk) {
            int k0 = kk*4 + 2*half;                  // A: lanes<16 K0/K1, >=16 K2/K3
            v2f a, bf;
            a.x  = buf[m0 + r][k0];
            a.y  = buf[m0 + r][k0 + 1];
            bf.x = buf[n0 + r][k0];                  // B(k,n) = centered[n][k]
            bf.y = buf[n0 + r][k0 + 1];
            acc = __builtin_amdgcn_wmma_f32_16x16x4_f32(
                      false, a, false, bf, (short)0, acc, false, false);
        }
        __syncthreads();                             // done before buffer reuse
    }
#else
    // Fallback: cooperative VMEM staging (toolchain without TDM builtin)
    int lrow = tid >> 3;
    int lk   = (tid & 7) * 8;
    const float* rowp = base + (size_t)lrow * TEXTN;
    v8f acc = {};
    for (int c = 0; c < NCHUNK; ++c) {
        int kbase = c * 64;
        __syncthreads();
#pragma unroll
        for (int u = 0; u < 8; ++u) {
            int k = kbase + lk + u;
            sh[0][lrow][lk + u] = (k < TT) ? rowp[k] : 0.f;
        }
        __syncthreads();
#pragma unroll
        for (int kk = 0; kk < 16; ++kk) {
            int k0 = kk*4 + 2*half;
            v2f a, bf;
            a.x  = sh[0][m0 + r][k0];
            a.y  = sh[0][m0 + r][k0 + 1];
            bf.x = sh[0][n0 + r][k0];
            bf.y = sh[0][n0 + r][k0 + 1];
            acc = __builtin_amdgcn_wmma_f32_16x16x4_f32(
                      false, a, false, bf, (short)0, acc, false, false);
        }
    }
#endif

    // fused corr -> abs -> diag=1 -> 0.5*Asp + 0.5*Afunc epilogue
#pragma unroll
    for (int j = 0; j < 8; ++j) {
        int m = m0 + j + half*8;                  // C/D layout: M=j / j+8
        int n = n0 + r;
        float cov = acc[j];
        float ssm = ssBuf[chbase + m];
        float ssn = ssBuf[chbase + n];
        float den = sqrtf(ssm * ssn);
        float rr  = (den > 0.f) ? cov / fmaxf(den, 1e-30f) : 0.f;
        if (m == n) rr = 1.f;
        float val = fabsf(rr);
        int gi = f*NCC + m, gj = f*NCC + n;
        Aout[(size_t)bb*NN*NN + (size_t)gi*NN + gj] =
            0.5f*Asp[(size_t)gi*NN + gj] + 0.5f*val;
    }
}

// -------------------- degree + normalization -------------------------
__global__ void rowsum_dinv(const float* __restrict__ Aout, float* __restrict__ dinv)
{
    int row = blockIdx.x*blockDim.x + threadIdx.x;   // 0..NB*NN-1
    if (row >= NB*NN) return;
    const float* p = Aout + (size_t)row * NN;
    float D = 0.f;
    for (int j = 0; j < NN; ++j) D += p[j];
    dinv[row] = 1.0f / sqrtf(D + 1e-8f);
}

__global__ void norm_A(const float* __restrict__ dinv, float* __restrict__ Aout)
{
    int idx = blockIdx.x*blockDim.x + threadIdx.x;
    const int total = NB*NN*NN;
    if (idx >= total) return;
    int b   = idx / (NN*NN);
    int rem = idx - b*(NN*NN);
    int i   = rem / NN;
    int jj  = rem - i*NN;
    Aout[idx] *= dinv[b*NN + i] * dinv[b*NN + jj];
}

extern "C" void kernel_launch(void* const* d_in, const int* in_sizes, int n_in,
                              void* d_out, int out_size, void* d_ws, size_t ws_size,
                              hipStream_t stream) {
    const float* eeg = (const float*)d_in[0];
    const float* bco = (const float*)d_in[1];
    const float* aco = (const float*)d_in[2];
    const float* zic = (const float*)d_in[3];
    const float* Asp = (const float*)d_in[4];

    float* out   = (float*)d_out;
    float* Aout  = out;                               // (B, 320, 320)
    float* nodef = out + (size_t)NB*NN*NN;            // (B, 320, 1)
    float* bandp = nodef + (size_t)NB*NN;             // (B, 64, 5)

    float* filtExt = (float*)d_ws;                    // NCH * 4054 floats (~166 MB)
    float* ssBuf   = filtExt + (size_t)NCH*TEXTN;
    float* dinv    = ssBuf + NCH;

    filt_forward <<<(NCH+255)/256, 256, 0, stream>>>(eeg, bco, aco, zic, filtExt);
    filt_backward<<<(NCH+255)/256, 256, 0, stream>>>(bco, aco, zic, filtExt,
                                                     ssBuf, nodef, bandp);
    const int totA = NB*NN*NN;
    init_A   <<<(totA+255)/256, 256, 0, stream>>>(Asp, Aout);
    cov_wmma <<<NFB*NB, 512, 0, stream>>>(filtExt, ssBuf, Asp, Aout);
    rowsum_dinv<<<(NB*NN+255)/256, 256, 0, stream>>>(Aout, dinv);
    norm_A   <<<(totA+255)/256, 256, 0, stream>>>(dinv, Aout);
}